// ContinuousGaussianCRFConv_72662256714586
// MI455X (gfx1250) — compile-verified
//
#include <hip/hip_runtime.h>
#include <hip/hip_bf16.h>

// ---------------------------------------------------------------------------
// ContinuousGaussianCRFConv for MI455X (gfx1250, wave32, WMMA)
//
// All dense GEMMs run through V_WMMA_F32_16X16X4_F32 (fp32 to match the
// reference numerics). The per-block B slice (weights) is staged in LDS so
// the hot loop is: 1x global_load_b64 (A frag) + 1x ds_load_b64 (B frag) +
// 1x v_wmma per k-step per matrix. Edge softmax / scatter phases are
// L2-resident (per-node H=64 state is 25.6MB << 192MB L2).
// ---------------------------------------------------------------------------

typedef __attribute__((ext_vector_type(2))) float v2f;
typedef __attribute__((ext_vector_type(8))) float v8f;

#define DU 256
#define DP 256
#define HC 64
#define DO 256
#define BN_EPS 1e-5f
#define LEAKY 0.01f

#define KMAX 256
#define KPAD (KMAX + 2)   // LDS row pad: keeps ds_load_b64 8B-aligned, spreads banks
#define WAVES_PER_BLOCK 8

// ---------------------------------------------------------------------------
// fp32 WMMA GEMM: Out[nRows x Dout] = A0[nRows x K] * B0[K x Dout]
//                                  (+ A1 * B1 when DUAL)
// Grid: x = row-tile strips (8 row tiles / block), y = column tile.
// All 8 waves of a block share one 16-column B slice staged in LDS
// (transposed to [n][k] so each lane's B fragment is one aligned b64 read).
// Optional fused BatchNorm statistics epilogue: every lane's 8 accumulator
// elements live in a single output column n, so per-lane partial sum /
// sum-of-squares need only 2 global atomics.
//
// fp32 WMMA fragment layout (ISA 7.12.2):
//   A (16x4):  lane = (k>=2)*16 + m ; vgpr v in {0,1} -> k = 2*(lane>>4)+v
//   B (4x16):  lane = (k>=2)*16 + n ; vgpr v in {0,1} -> k = 2*(lane>>4)+v
//   C/D(16x16): lanes 0-15: M=v, N=lane ; lanes 16-31: M=v+8, N=lane-16
// ---------------------------------------------------------------------------
template <bool DUAL, bool STATS>
__global__ void wmma_gemm_f32_kernel(const float* __restrict__ A0,
                                     const float* __restrict__ B0,
                                     const float* __restrict__ A1,
                                     const float* __restrict__ B1,
                                     float* __restrict__ Out,
                                     float* __restrict__ bnSum,
                                     float* __restrict__ bnSumSq,
                                     int nRows, int K, int Dout) {
  __shared__ float ldsB[(DUAL ? 2 : 1) * 16 * KPAD];

  const int tid = threadIdx.x;
  const int colBase = blockIdx.y * 16;

  // Stage this block's 16-column B slice(s): global [k][n] -> LDS [n][k].
  for (int idx = tid; idx < K * 16; idx += blockDim.x) {
    int k = idx >> 4, n = idx & 15;
    ldsB[n * KPAD + k] = B0[(size_t)k * Dout + colBase + n];
    if (DUAL) ldsB[16 * KPAD + n * KPAD + k] = B1[(size_t)k * Dout + colBase + n];
  }
  __syncthreads();

  const int lane = tid & 31;
  const int rowTile = blockIdx.x * WAVES_PER_BLOCK + (tid >> 5);
  if (rowTile * 16 >= nRows) return;  // wave-uniform: EXEC stays all-ones

  const int m  = lane & 15;         // row (A) / column (B) within the tile
  const int kq = (lane >> 4) << 1;  // 0 or 2: which k-pair this half-wave holds

  const float* a0row = A0 + (size_t)(rowTile * 16 + m) * K + kq;
  const float* a1row = DUAL ? (A1 + (size_t)(rowTile * 16 + m) * K + kq) : nullptr;
  const float* bl0 = &ldsB[m * KPAD + kq];
  const float* bl1 = DUAL ? &ldsB[16 * KPAD + m * KPAD + kq] : nullptr;

  v8f acc = {};
#pragma unroll 4
  for (int k = 0; k < K; k += 4) {
    v2f a = *(const v2f*)(a0row + k);   // 8B-aligned: one global_load_b64
    v2f b = *(const v2f*)(bl0 + k);     // 8B-aligned: one ds_load_b64
    acc = __builtin_amdgcn_wmma_f32_16x16x4_f32(false, a, false, b,
                                                (short)0, acc, false, false);
    if (DUAL) {
      v2f a1 = *(const v2f*)(a1row + k);
      v2f b1 = *(const v2f*)(bl1 + k);
      acc = __builtin_amdgcn_wmma_f32_16x16x4_f32(false, a1, false, b1,
                                                  (short)0, acc, false, false);
    }
  }

  const int n = colBase + m;
  const int rowOff = (lane >> 4) * 8;
  const size_t outBase = (size_t)(rowTile * 16) * Dout + n;
  float s = 0.f, s2 = 0.f;
#pragma unroll
  for (int v = 0; v < 8; ++v) {
    float val = acc[v];
    Out[outBase + (size_t)(v + rowOff) * Dout] = val;
    s += val;
    s2 += val * val;
  }
  if (STATS) {  // fused BN-statistics epilogue (column-uniform per lane)
    atomicAdd(bnSum + n, s);
    atomicAdd(bnSumSq + n, s2);
  }
}

// ---------------------------------------------------------------------------
// BatchNorm: fold batch stats into per-channel scale/shift.
// ---------------------------------------------------------------------------
__global__ void bn_finalize_kernel(const float* __restrict__ sum,
                                   const float* __restrict__ sumsq,
                                   const float* __restrict__ gamma,
                                   const float* __restrict__ beta,
                                   float* __restrict__ ss, int D, float invN) {
  int ch = blockIdx.x * blockDim.x + threadIdx.x;
  if (ch >= D) return;
  float mu  = sum[ch] * invN;
  float var = sumsq[ch] * invN - mu * mu;  // biased variance (torch/jnp.var)
  float sc  = gamma[ch] * rsqrtf(var + BN_EPS);
  ss[ch]     = sc;
  ss[D + ch] = beta[ch] - mu * sc;
}

__global__ void bn_apply_kernel(float* __restrict__ data,
                                const float* __restrict__ ss,
                                long long total, int Dmask, int D, int leaky) {
  long long t = (long long)blockIdx.x * blockDim.x + threadIdx.x;
  if (t >= total) return;
  int ch = (int)(t & Dmask);  // D is a power of two
  float v = data[t] * ss[ch] + ss[D + ch];
  if (leaky) v = v > 0.f ? v : LEAKY * v;
  data[t] = v;
}

// ---------------------------------------------------------------------------
// Workspace init: zero accumulators, seed segment-max encodings.
// ---------------------------------------------------------------------------
__global__ void init_ws_kernel(float* __restrict__ agg, float* __restrict__ denom,
                               unsigned* __restrict__ menc, float* __restrict__ sums,
                               long long nAgg, int nN, int nSums) {
  long long t = (long long)blockIdx.x * blockDim.x + threadIdx.x;
  if (t < nAgg) agg[t] = 0.f;
  if (t < nN) { denom[t] = 0.f; menc[t] = 0u; }
  if (t < nSums) sums[t] = 0.f;
}

// Order-preserving float <-> uint encoding for atomicMax-based segment_max.
__device__ __forceinline__ unsigned enc_f32(float x) {
  unsigned u = __float_as_uint(x);
  return (u & 0x80000000u) ? ~u : (u | 0x80000000u);
}
__device__ __forceinline__ float dec_f32(unsigned e) {
  unsigned u = (e & 0x80000000u) ? (e & 0x7FFFFFFFu) : ~e;
  return __uint_as_float(u);
}

// Pass 1: logit[e] = -||su[i]-su[j]||^2 ; segment max by destination i.
__global__ void edge_logit_kernel(const float* __restrict__ su,
                                  const long long* __restrict__ ei,
                                  float* __restrict__ logit,
                                  unsigned* __restrict__ menc, int E_) {
  int e = blockIdx.x * blockDim.x + threadIdx.x;
  if (e >= E_) return;
  long long i = ei[e];
  long long j = ei[(size_t)E_ + e];
  const float4* si = (const float4*)(su + i * HC);
  const float4* sj = (const float4*)(su + j * HC);
  float acc = 0.f;
#pragma unroll
  for (int q = 0; q < HC / 4; ++q) {
    float4 a = si[q], b = sj[q];
    float dx = a.x - b.x, dy = a.y - b.y, dz = a.z - b.z, dw = a.w - b.w;
    acc += dx * dx + dy * dy + dz * dz + dw * dw;
  }
  float lg = -acc;
  logit[e] = lg;
  atomicMax(menc + i, enc_f32(lg));
}

// Pass 2: ex = exp(logit - max_i) ; segment sum of ex by i.
__global__ void edge_exp_kernel(const float* __restrict__ logit,
                                const long long* __restrict__ ei,
                                const unsigned* __restrict__ menc,
                                float* __restrict__ denom,
                                float* __restrict__ ex, int E_) {
  int e = blockIdx.x * blockDim.x + threadIdx.x;
  if (e >= E_) return;
  long long i = ei[e];
  float mx = dec_f32(menc[i]);
  float v = expf(logit[e] - mx);
  ex[e] = v;
  atomicAdd(denom + i, v);
}

// Pass 3: agg[i] += (ex/denom[i]) * xu[j]  (h == z == xu for STEPS==1).
// Two threads per edge, 32 channels each; gathers are L2-resident.
__global__ void edge_agg_kernel(const float* __restrict__ ex,
                                const float* __restrict__ denom,
                                const long long* __restrict__ ei,
                                const float* __restrict__ xu,
                                float* __restrict__ agg, int E_) {
  int t = blockIdx.x * blockDim.x + threadIdx.x;
  int e = t >> 1;
  if (e >= E_) return;
  int half = t & 1;
  long long i = ei[e];
  long long j = ei[(size_t)E_ + e];
  float w = ex[e] / (denom[i] + 1e-16f);
  const float* src = xu + j * HC + half * 32;
  float* dst = agg + i * HC + half * 32;
#pragma unroll
  for (int q = 0; q < 32; ++q) atomicAdd(dst + q, w * src[q]);
}

// ---------------------------------------------------------------------------
// Single block: CC = c^T c ; Minv = inv(I + CC) (SPD -> pivot-free
// Gauss-Jordan in LDS) ; M2 = CC * Minv.  h = xu*Minv + agg*M2 afterwards.
// ---------------------------------------------------------------------------
__global__ void build_mats_kernel(const float* __restrict__ c,
                                  float* __restrict__ CC,
                                  float* __restrict__ Minv,
                                  float* __restrict__ M2) {
  __shared__ float aug[HC][2 * HC + 2];  // 64 x 130 floats = 33.3 KB LDS
  const int tid = threadIdx.x;           // 256 threads

  for (int idx = tid; idx < HC * HC; idx += 256) {
    int a = idx >> 6, b = idx & 63;
    float s = 0.f;
    for (int k = 0; k < HC; ++k) s += c[k * HC + a] * c[k * HC + b];
    CC[idx] = s;
  }
  __syncthreads();
  for (int idx = tid; idx < HC * HC; idx += 256) {
    int r = idx >> 6, col = idx & 63;
    aug[r][col] = CC[idx] + (r == col ? 1.f : 0.f);
    aug[r][HC + col] = (r == col ? 1.f : 0.f);
  }
  __syncthreads();

  const int r  = tid >> 2;         // row this thread eliminates
  const int c0 = (tid & 3) * 32;   // 32-column slice
  for (int p = 0; p < HC; ++p) {
    float pivinv = 1.0f / aug[p][p];
    __syncthreads();
    if (tid < 2 * HC) aug[p][tid] *= pivinv;
    __syncthreads();
    float f = aug[r][p];
    __syncthreads();
    if (r != p) {
      for (int cc = c0; cc < c0 + 32; ++cc) aug[r][cc] -= f * aug[p][cc];
    }
    __syncthreads();
  }

  for (int idx = tid; idx < HC * HC; idx += 256) {
    int rr = idx >> 6, col = idx & 63;
    Minv[idx] = aug[rr][HC + col];
  }
  __syncthreads();
  for (int idx = tid; idx < HC * HC; idx += 256) {
    int a = idx >> 6, b = idx & 63;
    float s = 0.f;
    for (int k = 0; k < HC; ++k) s += CC[a * HC + k] * aug[k][HC + b];
    M2[idx] = s;
  }
}

// ---------------------------------------------------------------------------
// Host-side launch orchestration (graph-capture safe: kernels only).
// ---------------------------------------------------------------------------
static inline int cdiv(long long a, long long b) { return (int)((a + b - 1) / b); }

extern "C" void kernel_launch(void* const* d_in, const int* in_sizes, int n_in,
                              void* d_out, int out_size, void* d_ws, size_t ws_size,
                              hipStream_t stream) {
  const float* x   = (const float*)d_in[0];
  const float* y   = (const float*)d_in[1];
  // d_in[2] = pos (unused by the reference computation)
  const long long* ei = (const long long*)d_in[3];  // int64 edge_index [2, E]
  const float* Wu  = (const float*)d_in[4];
  const float* gu  = (const float*)d_in[5];
  const float* bu  = (const float*)d_in[6];
  const float* Wp  = (const float*)d_in[7];
  const float* gp  = (const float*)d_in[8];
  const float* bp  = (const float*)d_in[9];
  const float* cM  = (const float*)d_in[10];
  const float* Wm  = (const float*)d_in[11];
  const float* gm  = (const float*)d_in[12];
  const float* bm  = (const float*)d_in[13];
  const float* Wf  = (const float*)d_in[14];
  const float* gf  = (const float*)d_in[15];
  const float* bf  = (const float*)d_in[16];
  float* out = (float*)d_out;

  const int N = in_sizes[0] / DU;   // 100000 (multiple of 16 -> no GEMM tails)
  const int E = in_sizes[3] / 2;    // 3200000

  // ---- workspace carve-up (fp32 slots) ----
  float* ws = (float*)d_ws;
  size_t o = 0;
  float*    xu    = ws + o; o += (size_t)N * HC;
  float*    su    = ws + o; o += (size_t)N * HC;
  float*    agg   = ws + o; o += (size_t)N * HC;
  float*    hbuf  = ws + o; o += (size_t)N * HC;
  float*    hm    = ws + o; o += (size_t)N * DO;
  float*    logit = ws + o; o += (size_t)E;
  float*    exw   = ws + o; o += (size_t)E;
  unsigned* menc  = (unsigned*)(ws + o); o += (size_t)N;
  float*    denom = ws + o; o += (size_t)N;
  float*    sum_u = ws + o; o += HC;   float* sq_u = ws + o; o += HC;
  float*    sum_p = ws + o; o += HC;   float* sq_p = ws + o; o += HC;
  float*    sum_m = ws + o; o += DO;   float* sq_m = ws + o; o += DO;
  float*    sum_f = ws + o; o += DO;   float* sq_f = ws + o; o += DO;
  float*    ss_u  = ws + o; o += 2 * HC;
  float*    ss_p  = ws + o; o += 2 * HC;
  float*    ss_m  = ws + o; o += 2 * DO;
  float*    ss_f  = ws + o; o += 2 * DO;
  float*    CC    = ws + o; o += HC * HC;
  float*    Minv  = ws + o; o += HC * HC;
  float*    M2    = ws + o; o += HC * HC;
  (void)ws_size; (void)n_in; (void)out_size;

  const int nSums = 2 * (HC + HC + DO + DO);  // contiguous sum/sq block
  const long long nAgg = (long long)N * HC;
  const float invN = 1.0f / (float)N;
  const int rowStrips = cdiv(N / 16, WAVES_PER_BLOCK);

  // 0) init accumulators (agg, denom, segment-max seeds, BN sums)
  init_ws_kernel<<<cdiv(nAgg, 256), 256, 0, stream>>>(agg, denom, menc, sum_u,
                                                      nAgg, N, nSums);

  // 1) xu_pre = x @ Wu ; su_pre = y @ Wp   (WMMA, fused BN stats)
  {
    dim3 grid(rowStrips, HC / 16);
    wmma_gemm_f32_kernel<false, true><<<grid, 256, 0, stream>>>(
        x, Wu, nullptr, nullptr, xu, sum_u, sq_u, N, DU, HC);
    wmma_gemm_f32_kernel<false, true><<<grid, 256, 0, stream>>>(
        y, Wp, nullptr, nullptr, su, sum_p, sq_p, N, DP, HC);
  }
  // 2) BN finalize + apply
  bn_finalize_kernel<<<1, HC, 0, stream>>>(sum_u, sq_u, gu, bu, ss_u, HC, invN);
  bn_finalize_kernel<<<1, HC, 0, stream>>>(sum_p, sq_p, gp, bp, ss_p, HC, invN);
  bn_apply_kernel<<<cdiv(nAgg, 256), 256, 0, stream>>>(xu, ss_u, nAgg, HC - 1, HC, 0);
  bn_apply_kernel<<<cdiv(nAgg, 256), 256, 0, stream>>>(su, ss_p, nAgg, HC - 1, HC, 0);

  // 3) edge softmax (3 passes, L2-resident gathers)
  edge_logit_kernel<<<cdiv(E, 256), 256, 0, stream>>>(su, ei, logit, menc, E);
  edge_exp_kernel<<<cdiv(E, 256), 256, 0, stream>>>(logit, ei, menc, denom, exw, E);

  // 4) CC = c^T c ; Minv = inv(I+CC) ; M2 = CC*Minv   (single block)
  build_mats_kernel<<<1, 256, 0, stream>>>(cM, CC, Minv, M2);

  // 5) agg[i] += softmax_w * xu[j]
  edge_agg_kernel<<<cdiv((long long)E * 2, 256), 256, 0, stream>>>(exw, denom, ei,
                                                                   xu, agg, E);

  // 6) h = xu @ Minv + agg @ M2   (dual WMMA GEMM)
  {
    dim3 grid(rowStrips, HC / 16);
    wmma_gemm_f32_kernel<true, false><<<grid, 256, 0, stream>>>(
        xu, Minv, agg, M2, hbuf, nullptr, nullptr, N, HC, HC);
  }

  // 7) hm_pre = h @ Wm  (WMMA + BN stats) ; BN + leaky ReLU
  {
    dim3 grid(rowStrips, DO / 16);
    wmma_gemm_f32_kernel<false, true><<<grid, 256, 0, stream>>>(
        hbuf, Wm, nullptr, nullptr, hm, sum_m, sq_m, N, HC, DO);
  }
  bn_finalize_kernel<<<1, DO, 0, stream>>>(sum_m, sq_m, gm, bm, ss_m, DO, invN);
  bn_apply_kernel<<<cdiv((long long)N * DO, 256), 256, 0, stream>>>(
      hm, ss_m, (long long)N * DO, DO - 1, DO, 1);

  // 8) out_pre = cat[hm, y] @ Wf = hm @ Wf[0:256] + y @ Wf[256:512]
  {
    dim3 grid(rowStrips, DO / 16);
    wmma_gemm_f32_kernel<true, true><<<grid, 256, 0, stream>>>(
        hm, Wf, y, Wf + (size_t)DO * DO, out, sum_f, sq_f, N, DO, DO);
  }
  bn_finalize_kernel<<<1, DO, 0, stream>>>(sum_f, sq_f, gf, bf, ss_f, DO, invN);
  bn_apply_kernel<<<cdiv((long long)N * DO, 256), 256, 0, stream>>>(
      out, ss_f, (long long)N * DO, DO - 1, DO, 1);
}